// LearnedRouter_84765474554513
// MI455X (gfx1250) — compile-verified
//
#include <hip/hip_runtime.h>
#include <hip/hip_bf16.h>

typedef float v2f __attribute__((ext_vector_type(2)));
typedef float v8f __attribute__((ext_vector_type(8)));

#define TOKENS   (8 * 4096)     // 32768
#define KDIM     2048
#define NEXPERT  64
#define TOPK     8

// ---------------------------------------------------------------------------
// Kernel 1: logits = x @ W^T via V_WMMA_F32_16X16X4_F32 (fp32 matrix path).
// One wave handles a 32-token x 64-expert block: 2 M-tiles x 4 N-tiles =
// 8 accumulator tiles, so each K-step issues 6 loads (2 A + 4 B shared by
// both M-tiles) for 8 WMMAs -> WMMA-issue bound rather than L1-load bound.
//
// A fragment (16x4 f32): lanes 0-15 hold M=lane, K={k,k+1}; lanes 16-31 hold
// M=lane-16, K={k+2,k+3}  -> contiguous float2 from row-major x.
// B fragment (4x16 f32): lanes 0-15 hold N=lane, K={k,k+1}; lanes 16-31 hold
// N=lane-16, K={k+2,k+3}. B = W^T so B[k][n] = W[n][k] -> contiguous float2
// from row-major W. Both loads are perfectly coalesced (256 B / fragment).
// ---------------------------------------------------------------------------
#define GEMM_WAVES 4
#define M_PER_WAVE 32           // 2 M-tiles of 16 tokens

__global__ __launch_bounds__(GEMM_WAVES * 32)
void router_gemm_kernel(const float* __restrict__ x,
                        const float* __restrict__ W,
                        float* __restrict__ logits) {
    const int lane  = threadIdx.x & 31;
    const int wave  = threadIdx.x >> 5;
    const int mBase = (blockIdx.x * GEMM_WAVES + wave) * M_PER_WAVE;
    const int half  = lane >> 4;                        // 0: lanes 0-15, 1: lanes 16-31
    const int sub   = lane & 15;
    const int koff  = half << 1;                        // 0 or 2

    const float* xp0 = x + (size_t)(mBase + sub)      * KDIM + koff;
    const float* xp1 = x + (size_t)(mBase + 16 + sub) * KDIM + koff;   // +131072 B
    const float* w0  = W + (size_t)(sub +  0) * KDIM + koff;
    const float* w1  = W + (size_t)(sub + 16) * KDIM + koff;
    const float* w2  = W + (size_t)(sub + 32) * KDIM + koff;
    const float* w3  = W + (size_t)(sub + 48) * KDIM + koff;

    v8f acc00 = {}, acc01 = {}, acc02 = {}, acc03 = {};   // M-tile 0, N-tiles 0..3
    v8f acc10 = {}, acc11 = {}, acc12 = {}, acc13 = {};   // M-tile 1, N-tiles 0..3

#pragma unroll 2
    for (int k = 0; k < KDIM; k += 4) {
        v2f a0 = *(const v2f*)(xp0 + k);
        v2f a1 = *(const v2f*)(xp1 + k);
        v2f b0 = *(const v2f*)(w0 + k);
        v2f b1 = *(const v2f*)(w1 + k);
        v2f b2 = *(const v2f*)(w2 + k);
        v2f b3 = *(const v2f*)(w3 + k);
        acc00 = __builtin_amdgcn_wmma_f32_16x16x4_f32(false, a0, false, b0,
                                                      (short)0, acc00, false, false);
        acc10 = __builtin_amdgcn_wmma_f32_16x16x4_f32(false, a1, false, b0,
                                                      (short)0, acc10, false, false);
        acc01 = __builtin_amdgcn_wmma_f32_16x16x4_f32(false, a0, false, b1,
                                                      (short)0, acc01, false, false);
        acc11 = __builtin_amdgcn_wmma_f32_16x16x4_f32(false, a1, false, b1,
                                                      (short)0, acc11, false, false);
        acc02 = __builtin_amdgcn_wmma_f32_16x16x4_f32(false, a0, false, b2,
                                                      (short)0, acc02, false, false);
        acc12 = __builtin_amdgcn_wmma_f32_16x16x4_f32(false, a1, false, b2,
                                                      (short)0, acc12, false, false);
        acc03 = __builtin_amdgcn_wmma_f32_16x16x4_f32(false, a0, false, b3,
                                                      (short)0, acc03, false, false);
        acc13 = __builtin_amdgcn_wmma_f32_16x16x4_f32(false, a1, false, b3,
                                                      (short)0, acc13, false, false);
    }

    // C/D layout: VGPR r -> M = r (lanes 0-15) or M = r+8 (lanes 16-31), N = sub.
    const int mAdd = half * 8;
#pragma unroll
    for (int r = 0; r < 8; ++r) {
        const size_t tok0 = (size_t)(mBase + mAdd + r);
        float* dst0 = logits + tok0 * NEXPERT + sub;
        dst0[ 0] = acc00[r];
        dst0[16] = acc01[r];
        dst0[32] = acc02[r];
        dst0[48] = acc03[r];
        float* dst1 = dst0 + (size_t)16 * NEXPERT;        // M-tile 1, +4096 B
        dst1[ 0] = acc10[r];
        dst1[16] = acc11[r];
        dst1[32] = acc12[r];
        dst1[48] = acc13[r];
    }
}

// ---------------------------------------------------------------------------
// Kernel 2: per-token softmax over 64 experts + top-8 (tie -> lowest index)
// + gate normalization. One wave32 per token, 2 experts per lane.
// ---------------------------------------------------------------------------
__global__ __launch_bounds__(256)
void router_topk_kernel(const float* __restrict__ logits,
                        float* __restrict__ idx_out,
                        float* __restrict__ probs_out,
                        float* __restrict__ gate_out) {
    const int lane  = threadIdx.x & 31;
    const int token = blockIdx.x * (blockDim.x >> 5) + (threadIdx.x >> 5);
    if (token >= TOKENS) return;

    const float* lrow = logits + (size_t)token * NEXPERT;
    float l0 = lrow[lane];
    float l1 = lrow[lane + 32];

    // wave max
    float m = fmaxf(l0, l1);
#pragma unroll
    for (int s = 16; s > 0; s >>= 1) m = fmaxf(m, __shfl_xor(m, s, 32));

    float e0 = __expf(l0 - m);
    float e1 = __expf(l1 - m);
    float sum = e0 + e1;
#pragma unroll
    for (int s = 16; s > 0; s >>= 1) sum += __shfl_xor(sum, s, 32);

    const float p0 = e0 / sum;
    const float p1 = e1 / sum;
    probs_out[(size_t)token * NEXPERT + lane]      = p0;
    probs_out[(size_t)token * NEXPERT + lane + 32] = p1;

    // iterative top-8 argmax; ties resolved toward the lower expert index
    float c0 = p0, c1 = p1;
    float myGate = 0.0f;
    int   myIdx  = 0;
    float gsum   = 0.0f;

#pragma unroll
    for (int i = 0; i < TOPK; ++i) {
        float v;
        int   id;
        if (c0 >= c1) { v = c0; id = lane; }          // lane < lane+32: ties -> c0
        else          { v = c1; id = lane + 32; }
#pragma unroll
        for (int s = 16; s > 0; s >>= 1) {
            float ov = __shfl_xor(v, s, 32);
            int   oi = __shfl_xor(id, s, 32);
            if (ov > v || (ov == v && oi < id)) { v = ov; id = oi; }
        }
        gsum += v;                                    // uniform across lanes
        if (lane == i) { myGate = v; myIdx = id; }
        if (id == lane)           c0 = -__builtin_inff();
        else if (id == lane + 32) c1 = -__builtin_inff();
    }

    if (lane < TOPK) {
        idx_out[(size_t)token * TOPK + lane]  = (float)myIdx;
        gate_out[(size_t)token * TOPK + lane] = myGate / gsum;
    }
}

// ---------------------------------------------------------------------------
extern "C" void kernel_launch(void* const* d_in, const int* in_sizes, int n_in,
                              void* d_out, int out_size, void* d_ws, size_t ws_size,
                              hipStream_t stream) {
    const float* x = (const float*)d_in[0];   // [8, 4096, 2048]
    const float* W = (const float*)d_in[1];   // [64, 2048]

    // d_out layout (floats), tuple return order:
    //   topk_idx [T,8] | probs [T,64] | gate [T,8] | logits [T,64]
    float* out      = (float*)d_out;
    float* o_idx    = out;
    float* o_probs  = out + (size_t)TOKENS * TOPK;
    float* o_gate   = o_probs + (size_t)TOKENS * NEXPERT;
    float* o_logits = o_gate + (size_t)TOKENS * TOPK;

    // GEMM: 32768/32 = 1024 wave-blocks of 32 tokens, 4 waves per block.
    const int waveTiles  = TOKENS / M_PER_WAVE;         // 1024
    const int gemmBlocks = waveTiles / GEMM_WAVES;      // 256
    router_gemm_kernel<<<gemmBlocks, GEMM_WAVES * 32, 0, stream>>>(x, W, o_logits);

    // Softmax + top-k: one wave per token, 8 waves per block.
    const int tkBlocks = TOKENS / 8;                    // 4096
    router_topk_kernel<<<tkBlocks, 256, 0, stream>>>(o_logits, o_idx, o_probs, o_gate);
}